// MNIST_CNN_52012053955144
// MI455X (gfx1250) — compile-verified
//
#include <hip/hip_runtime.h>
#include <math.h>

typedef __attribute__((ext_vector_type(2))) float v2f;
typedef __attribute__((ext_vector_type(8))) float v8f;
typedef unsigned int u32x4 __attribute__((ext_vector_type(4)));
typedef int i32x8 __attribute__((ext_vector_type(8)));
typedef int i32x4 __attribute__((ext_vector_type(4)));

// ---------------- workspace layout (float offsets) ----------------
static constexpr size_t O_S1    = 0;        // 1      : sum of per-image max|x|
static constexpr size_t O_PIMAX = 256;      // 2048   : per-image max|x|
static constexpr size_t O_W1    = 2304;     // 1600   : effective conv1 weights
static constexpr size_t O_B1    = 3904;     // 64     : effective conv1 bias
static constexpr size_t O_W2    = 4096;     // 102400 : effective conv2 weights [o][c][5][5]
static constexpr size_t O_B2    = 106496;   // 64
static constexpr size_t O_S3    = 106560;   // 1024   : per-feature sum |h2|
static constexpr size_t O_WL    = 107584;   // 10240  : effective linear weights [10][1024]
static constexpr size_t O_BL    = 117824;   // 16
static constexpr size_t O_PCMAX = 117840;   // 131072 : per-(img,ch) max of h1
static constexpr size_t O_H1    = 248912;   // 18874368 : (2048,64,12,12)
static constexpr size_t O_H2    = 19123280; // 2097152  : (2048,1024)
// total ~84.9 MB

__device__ __forceinline__ float scinol_p(float p0, float S2, float G, float eta,
                                          float M, bool cond) {
  float denom = sqrtf(S2 + M * M);
  float theta = fminf(fmaxf(G / denom, -1.f), 1.f);
  return p0 + (cond ? theta / (2.f * denom) * eta : 0.f);
}

// ---------------- stage 1: batch stat for conv1 ----------------
__global__ __launch_bounds__(256) void k_img_absmax(const float* __restrict__ x,
                                                    float* __restrict__ pimax) {
  __shared__ float red[256];
  int img = blockIdx.x, t = threadIdx.x;
  const float* p = x + (size_t)img * 784;
  float m = 0.f;
  for (int i = t; i < 784; i += 256) m = fmaxf(m, fabsf(p[i]));
  red[t] = m;
  __syncthreads();
  for (int s = 128; s > 0; s >>= 1) {
    if (t < s) red[t] = fmaxf(red[t], red[t + s]);
    __syncthreads();
  }
  if (t == 0) pimax[img] = red[0];
}

__global__ __launch_bounds__(256) void k_sum2048(const float* __restrict__ in,
                                                 float* __restrict__ out) {
  __shared__ float red[256];
  int t = threadIdx.x;
  float s = 0.f;
  for (int i = t; i < 2048; i += 256) s += in[i];
  red[t] = s;
  __syncthreads();
  for (int st = 128; st > 0; st >>= 1) {
    if (t < st) red[t] += red[t + st];
    __syncthreads();
  }
  if (t == 0) out[0] = red[0];
}

__global__ __launch_bounds__(256) void k_eff_w1(
    const float* __restrict__ s1,
    const float* __restrict__ w0, const float* __restrict__ wS2,
    const float* __restrict__ wG, const float* __restrict__ weta,
    const float* __restrict__ wM,
    const float* __restrict__ b0, const float* __restrict__ bS2,
    const float* __restrict__ bG, const float* __restrict__ beta,
    float* __restrict__ w1e, float* __restrict__ b1e) {
  float M = fmaxf(s1[0] * (1.f / 2048.f), wM[0]);
  for (int i = threadIdx.x; i < 1600; i += 256)
    w1e[i] = scinol_p(w0[i], wS2[i], wG[i], weta[i], M, wG[i] != 0.f);
  if (threadIdx.x < 64) {
    int j = threadIdx.x;
    b1e[j] = scinol_p(b0[j], bS2[j], bG[j], beta[j], 1.f, bG[j] != 0.f);
  }
}

// ---------------- conv1 + relu + pool + per-channel max (VALU; K=25,Cin=1) ---
__global__ __launch_bounds__(256) void k_conv1(
    const float* __restrict__ x, const float* __restrict__ w1e,
    const float* __restrict__ b1e, float* __restrict__ h1,
    float* __restrict__ pcmax) {
  __shared__ float sx[784];
  __shared__ float sw[1600];
  __shared__ float sb[64];
  __shared__ unsigned int smax[64];
  int img = blockIdx.x, t = threadIdx.x;
  const float* xin = x + (size_t)img * 784;
  for (int i = t; i < 784; i += 256) sx[i] = xin[i];
  for (int i = t; i < 1600; i += 256) sw[i] = w1e[i];
  if (t < 64) { sb[t] = b1e[t]; smax[t] = 0u; }
  __syncthreads();
  float* hout = h1 + (size_t)img * 9216;
  for (int o = t; o < 9216; o += 256) {
    int c = o / 144, pix = o % 144;
    int py = pix / 12, px = pix % 12;
    const float* wc = sw + c * 25;
    float best = 0.f;  // relu folded into the max
    #pragma unroll
    for (int dy = 0; dy < 2; ++dy)
      #pragma unroll
      for (int dx = 0; dx < 2; ++dx) {
        int iy = 2 * py + dy, ix = 2 * px + dx;
        float acc = sb[c];
        #pragma unroll
        for (int r = 0; r < 5; ++r)
          #pragma unroll
          for (int s = 0; s < 5; ++s)
            acc = fmaf(sx[(iy + r) * 28 + ix + s], wc[r * 5 + s], acc);
        best = fmaxf(best, acc);
      }
    hout[o] = best;
    atomicMax(&smax[c], __float_as_uint(best));  // best >= 0: bit-monotonic
  }
  __syncthreads();
  if (t < 64) pcmax[(size_t)img * 64 + t] = __uint_as_float(smax[t]);
}

// ---------------- conv2 stat + effective params ----------------
__global__ __launch_bounds__(256) void k_eff_w2(
    const float* __restrict__ pcmax, const float* __restrict__ wM,
    const float* __restrict__ w0, const float* __restrict__ wS2,
    const float* __restrict__ wG, const float* __restrict__ weta,
    const float* __restrict__ b0, const float* __restrict__ bS2,
    const float* __restrict__ bG, const float* __restrict__ beta,
    float* __restrict__ w2e, float* __restrict__ b2e) {
  __shared__ float red[256];
  __shared__ float Msh;
  int c = blockIdx.x, t = threadIdx.x;  // block per input channel
  float s = 0.f;
  for (int i = t; i < 2048; i += 256) s += pcmax[(size_t)i * 64 + c];
  red[t] = s;
  __syncthreads();
  for (int st = 128; st > 0; st >>= 1) {
    if (t < st) red[t] += red[t + st];
    __syncthreads();
  }
  if (t == 0) Msh = fmaxf(red[0] * (1.f / 2048.f), wM[c]);
  __syncthreads();
  float M = Msh;
  for (int j = t; j < 1600; j += 256) {  // all (o, r, s) for this input channel
    int o = j / 25, rs = j % 25;
    int idx = (o * 64 + c) * 25 + rs;
    w2e[idx] = scinol_p(w0[idx], wS2[idx], wG[idx], weta[idx], M, wG[idx] != 0.f);
  }
  if (c == 0 && t < 64)
    b2e[t] = scinol_p(b0[t], bS2[t], bG[t], beta[t], 1.f, bG[t] != 0.f);
}

// ---- TDM: async 2D tile load of a weight panel into LDS -------------------
// Tile: 64 rows (out-channels) x 100 DWORDs (K chunk), row stride 1600 DWORDs,
// packed contiguously into LDS -> exactly the [n][k_local] layout we want.
// D# per CDNA5 ISA (8.3/8.4): count=1, type=2 (image),
// data_size=4B, no multicast (not in a cluster), no pad/iterate.
// 6-arg toolchain form: (g0, g1, g2, g3, g4, cpol); groups 2..4 unused for 2D.
__device__ __forceinline__ void tdm_load_w_panel(const float* gsrc, float* ldsdst) {
  unsigned long long ga = (unsigned long long)(uintptr_t)gsrc;
  u32x4 g0;
  g0[0] = 1u;                                                    // count=1
  g0[1] = (unsigned)(uintptr_t)ldsdst;                           // LDS byte addr (addr[31:0])
  g0[2] = (unsigned)(ga & 0xFFFFFFFFull);                        // global_addr lo
  g0[3] = (unsigned)((ga >> 32) & 0x01FFFFFFull) | 0x80000000u;  // ga hi | type=2
  i32x8 g1;
  g1[0] = 0x00020000;          // data_size=2 (4B); workgroup_mask=0
  g1[1] = (int)(1600u << 16);  // tensor_dim0[15:0]=1600 (bits 79:48 start)
  g1[2] = (int)(64u << 16);    // tensor_dim0 hi=0 | tensor_dim1=64
  g1[3] = (int)(100u << 16);   // tensor_dim1 hi=0 | tile_dim0=100
  g1[4] = 64;                  // tile_dim1=64 | tile_dim2=0
  g1[5] = 1600;                // tensor_dim0_stride lo
  g1[6] = 0;                   // stride0 hi | stride1 lo
  g1[7] = 0;                   // stride1 hi
  i32x4 gz4 = {0, 0, 0, 0};
  i32x8 gz8 = {0, 0, 0, 0, 0, 0, 0, 0};
  __builtin_amdgcn_tensor_load_to_lds(g0, g1, gz4, gz4, gz8, 0);
}

// ---------------- conv2 as WMMA f32 GEMM, fused bias+relu+pool --------------
// Per image: A(im2col) = 64 pixels x 1600, B = 1600 x 64 channels.
// 8 waves x 2 tiles = 16 (16x16) C tiles; K chunked by 100 (4 channels).
// Per chunk: TDM fills sB (async) while all threads stage the im2col panel sA;
// inner loop is 3x ds_load_b64 + 2x v_wmma (stride-100 => conflict-free banks).
__global__ __launch_bounds__(256) void k_conv2_wmma(
    const float* __restrict__ h1, const float* __restrict__ w2e,
    const float* __restrict__ b2e, float* __restrict__ h2) {
  __shared__ float sA[64 * 100];  // 25.6 KB: [pixel m][k_local]
  __shared__ float sB[64 * 100];  // 25.6 KB: [channel n][k_local]
  int img = blockIdx.x, t = threadIdx.x;
  int wave = t >> 5, lane = t & 31;
  int l15 = lane & 15;
  int koff = (lane >= 16) ? 2 : 0;  // A/B element v <-> K = koff + v

  const float* hin = h1 + (size_t)img * 9216;

  // Chunk-invariant im2col gather offsets: element i = t + j*256 of sA maps to
  // pixel m = i/100, k_local kk = i%100 -> (c_local, r, s) tap.
  int soff[25];
  #pragma unroll
  for (int j = 0; j < 25; ++j) {
    int i = t + j * 256;
    int m = i / 100, kk = i % 100;
    int cl = kk / 25, rs = kk % 25;
    int r = rs / 5, s2 = rs % 5;
    int py = m >> 3, px = m & 7;
    soff[j] = cl * 144 + (py + r) * 12 + (px + s2);
  }

  int tile0 = wave, tile1 = wave + 8;  // tm = tile>>2 (pixel rows), tn = tile&3
  int tm0 = tile0 >> 2, tm1 = tile1 >> 2;
  int tn = tile0 & 3;                  // == tile1&3: both tiles share B
  int ia0 = (tm0 * 16 + l15) * 100 + koff;
  int ia1 = (tm1 * 16 + l15) * 100 + koff;
  int ib  = (tn  * 16 + l15) * 100 + koff;

  v8f acc0 = {}, acc1 = {};
  for (int chunk = 0; chunk < 16; ++chunk) {
    __syncthreads();  // previous chunk's sA/sB reads complete
    if (wave == 0) tdm_load_w_panel(w2e + chunk * 100, sB);  // async -> sB
    const float* hsrc = hin + chunk * 576;  // 4 input channels per chunk
    #pragma unroll
    for (int j = 0; j < 25; ++j) sA[t + j * 256] = hsrc[soff[j]];
    if (wave == 0) __builtin_amdgcn_s_wait_tensorcnt(0);
    __syncthreads();  // sA staged + TDM landed

    #pragma unroll 5
    for (int ks = 0; ks < 25; ++ks) {
      v2f a0 = *(const v2f*)&sA[ia0 + 4 * ks];
      v2f a1 = *(const v2f*)&sA[ia1 + 4 * ks];
      v2f bv = *(const v2f*)&sB[ib + 4 * ks];
      acc0 = __builtin_amdgcn_wmma_f32_16x16x4_f32(false, a0, false, bv,
                                                   (short)0, acc0, false, false);
      acc1 = __builtin_amdgcn_wmma_f32_16x16x4_f32(false, a1, false, bv,
                                                   (short)0, acc1, false, false);
    }
  }

  // Epilogue: bias + relu + 2x2 maxpool. Lane holds column N=l15, rows M=v(+8).
  // v == px at fixed py; lane l <-> l^16 is the py pair => shfl_xor(16).
  float* hout = h2 + (size_t)img * 1024;
  float bias = b2e[tn * 16 + l15];
  {
    float p[4];
    #pragma unroll
    for (int i = 0; i < 4; ++i) {
      float u0 = fmaxf(acc0[2 * i] + bias, 0.f);
      float u1 = fmaxf(acc0[2 * i + 1] + bias, 0.f);
      p[i] = fmaxf(u0, u1);
      p[i] = fmaxf(p[i], __shfl_xor(p[i], 16, 32));
    }
    if (lane < 16) {
      int n = tn * 16 + l15;
      #pragma unroll
      for (int i = 0; i < 4; ++i) hout[n * 16 + tm0 * 4 + i] = p[i];
    }
  }
  {
    float p[4];
    #pragma unroll
    for (int i = 0; i < 4; ++i) {
      float u0 = fmaxf(acc1[2 * i] + bias, 0.f);
      float u1 = fmaxf(acc1[2 * i + 1] + bias, 0.f);
      p[i] = fmaxf(u0, u1);
      p[i] = fmaxf(p[i], __shfl_xor(p[i], 16, 32));
    }
    if (lane < 16) {
      int n = tn * 16 + l15;
      #pragma unroll
      for (int i = 0; i < 4; ++i) hout[n * 16 + tm1 * 4 + i] = p[i];
    }
  }
}

// ---------------- linear stat + effective params ----------------
__global__ __launch_bounds__(256) void k_stat3(const float* __restrict__ h2,
                                               float* __restrict__ s3) {
  int f = blockIdx.x * 256 + threadIdx.x;  // 4 blocks x 256 = 1024 features
  float s = 0.f;
  for (int i = 0; i < 2048; ++i) s += fabsf(h2[(size_t)i * 1024 + f]);
  s3[f] = s;
}

__global__ __launch_bounds__(256) void k_eff_wl(
    const float* __restrict__ s3,
    const float* __restrict__ w0, const float* __restrict__ wS2,
    const float* __restrict__ wG, const float* __restrict__ weta,
    const float* __restrict__ wM,
    const float* __restrict__ b0, const float* __restrict__ bS2,
    const float* __restrict__ bG, const float* __restrict__ beta,
    float* __restrict__ wle, float* __restrict__ ble) {
  for (int i = threadIdx.x; i < 10240; i += 256) {
    int cidx = i & 1023;
    float M = fmaxf(wM[i], s3[cidx] * (1.f / 2048.f));
    wle[i] = scinol_p(w0[i], wS2[i], wG[i], weta[i], M, wS2[i] != 0.f);
  }
  if (threadIdx.x < 10) {
    int j = threadIdx.x;
    ble[j] = scinol_p(b0[j], bS2[j], bG[j], beta[j], 1.f, bS2[j] != 0.f);
  }
}

// ---------------- linear: 2048x10 = (2048x1024)@(1024x10), WMMA, N pad 16 ---
__global__ __launch_bounds__(128) void k_linear_wmma(
    const float* __restrict__ h2, const float* __restrict__ wle,
    const float* __restrict__ ble, float* __restrict__ out) {
  int wave = threadIdx.x >> 5, lane = threadIdx.x & 31;
  int l15 = lane & 15, koff = (lane >= 16) ? 2 : 0;
  int m0 = (blockIdx.x * 4 + wave) * 16;
  const float* arow = h2 + (size_t)(m0 + l15) * 1024;
  bool nvalid = l15 < 10;
  const float* brow = wle + (size_t)(nvalid ? l15 : 0) * 1024;
  v8f acc = {};
  for (int ks = 0; ks < 256; ++ks) {
    int k0 = ks * 4 + koff;
    v2f a, b;
    #pragma unroll
    for (int v = 0; v < 2; ++v) {
      a[v] = arow[k0 + v];
      b[v] = nvalid ? brow[k0 + v] : 0.f;
    }
    acc = __builtin_amdgcn_wmma_f32_16x16x4_f32(false, a, false, b,
                                                (short)0, acc, false, false);
  }
  if (nvalid) {
    float bias = ble[l15];
    #pragma unroll
    for (int v = 0; v < 8; ++v) {
      int row = m0 + v + ((lane < 16) ? 0 : 8);
      out[row * 10 + l15] = acc[v] + bias;
    }
  }
}

extern "C" void kernel_launch(void* const* d_in, const int* in_sizes, int n_in,
                              void* d_out, int out_size, void* d_ws, size_t ws_size,
                              hipStream_t stream) {
  (void)in_sizes; (void)n_in; (void)out_size; (void)ws_size;
  const float* x       = (const float*)d_in[0];
  const float* c1_w0   = (const float*)d_in[1];
  const float* c1_wS2  = (const float*)d_in[2];
  const float* c1_wG   = (const float*)d_in[3];
  const float* c1_weta = (const float*)d_in[4];
  const float* c1_wM   = (const float*)d_in[5];
  const float* c1_b0   = (const float*)d_in[6];
  const float* c1_bS2  = (const float*)d_in[7];
  const float* c1_bG   = (const float*)d_in[8];
  const float* c1_beta = (const float*)d_in[9];
  const float* c2_w0   = (const float*)d_in[10];
  const float* c2_wS2  = (const float*)d_in[11];
  const float* c2_wG   = (const float*)d_in[12];
  const float* c2_weta = (const float*)d_in[13];
  const float* c2_wM   = (const float*)d_in[14];
  const float* c2_b0   = (const float*)d_in[15];
  const float* c2_bS2  = (const float*)d_in[16];
  const float* c2_bG   = (const float*)d_in[17];
  const float* c2_beta = (const float*)d_in[18];
  const float* l_w0    = (const float*)d_in[19];
  const float* l_wS2   = (const float*)d_in[20];
  const float* l_wG    = (const float*)d_in[21];
  const float* l_weta  = (const float*)d_in[22];
  const float* l_wM    = (const float*)d_in[23];
  const float* l_b0    = (const float*)d_in[24];
  const float* l_bS2   = (const float*)d_in[25];
  const float* l_bG    = (const float*)d_in[26];
  const float* l_beta  = (const float*)d_in[27];

  float* ws    = (float*)d_ws;
  float* s1    = ws + O_S1;
  float* pimax = ws + O_PIMAX;
  float* w1e   = ws + O_W1;
  float* b1e   = ws + O_B1;
  float* w2e   = ws + O_W2;
  float* b2e   = ws + O_B2;
  float* s3    = ws + O_S3;
  float* wle   = ws + O_WL;
  float* ble   = ws + O_BL;
  float* pcmax = ws + O_PCMAX;
  float* h1    = ws + O_H1;
  float* h2    = ws + O_H2;

  k_img_absmax<<<2048, 256, 0, stream>>>(x, pimax);
  k_sum2048<<<1, 256, 0, stream>>>(pimax, s1);
  k_eff_w1<<<1, 256, 0, stream>>>(s1, c1_w0, c1_wS2, c1_wG, c1_weta, c1_wM,
                                  c1_b0, c1_bS2, c1_bG, c1_beta, w1e, b1e);
  k_conv1<<<2048, 256, 0, stream>>>(x, w1e, b1e, h1, pcmax);
  k_eff_w2<<<64, 256, 0, stream>>>(pcmax, c2_wM, c2_w0, c2_wS2, c2_wG, c2_weta,
                                   c2_b0, c2_bS2, c2_bG, c2_beta, w2e, b2e);
  k_conv2_wmma<<<2048, 256, 0, stream>>>(h1, w2e, b2e, h2);
  k_stat3<<<4, 256, 0, stream>>>(h2, s3);
  k_eff_wl<<<1, 256, 0, stream>>>(s3, l_w0, l_wS2, l_wG, l_weta, l_wM,
                                  l_b0, l_bS2, l_bG, l_beta, wle, ble);
  k_linear_wmma<<<32, 128, 0, stream>>>(h2, wle, ble, (float*)d_out);
}